// GCNNet_71734543778230
// MI455X (gfx1250) — compile-verified
//
#include <hip/hip_runtime.h>
#include <math.h>
#include <stdint.h>

typedef __attribute__((ext_vector_type(2))) float v2f;
typedef __attribute__((ext_vector_type(8))) float v8f;

__device__ __forceinline__ v8f wmma_f32_16x16x4(v2f a, v2f b, v8f c) {
    // D = A(16x4 f32) * B(4x16 f32) + C(16x16 f32); wave32, full fp32 precision.
    return __builtin_amdgcn_wmma_f32_16x16x4_f32(
        /*neg_a=*/false, a, /*neg_b=*/false, b,
        /*c_mod=*/(short)0, c, /*reuse_a=*/false, /*reuse_b=*/false);
}

__device__ __forceinline__ void wait_asynccnt0() {
#if __has_builtin(__builtin_amdgcn_s_wait_asynccnt)
    __builtin_amdgcn_s_wait_asynccnt(0);
#else
    asm volatile("s_wait_asynccnt 0" ::: "memory");
#endif
}

// Async global->LDS copy of 16 bytes per lane (GLOBAL_LOAD_ASYNC_TO_LDS_B128,
// GVS addressing: saddr = uniform base, vaddr = per-lane byte offset).
__device__ __forceinline__ void async_copy_b128(unsigned lds_off, unsigned gbl_off,
                                                const float* base) {
    asm volatile("global_load_async_to_lds_b128 %0, %1, %2"
                 :: "v"(lds_off), "v"(gbl_off), "s"(base)
                 : "memory");
}

// ---------------------------------------------------------------------------
// Utility kernels
// ---------------------------------------------------------------------------
__global__ void zero_f32(float* __restrict__ p, long long n) {
    long long i = (long long)blockIdx.x * blockDim.x + threadIdx.x;
    if (i < n) p[i] = 0.0f;
}

__global__ void count_deg(const int* __restrict__ dst, float* __restrict__ deg, int E) {
    int e = blockIdx.x * blockDim.x + threadIdx.x;
    if (e < E) unsafeAtomicAdd(&deg[dst[e]], 1.0f);
}

__global__ void compute_dis(const float* __restrict__ deg, float* __restrict__ dis, int N) {
    int i = blockIdx.x * blockDim.x + threadIdx.x;
    if (i < N) {
        // self-loop contributes +1, so total degree >= 1 > 0 always
        dis[i] = rsqrtf(deg[i] + 1.0f);
    }
}

// ---------------------------------------------------------------------------
// GEMM1: C[M,ncols] = A[M,K] @ B[K,ncols]   (M%64==0, K%32==0, ncols<=48)
// 128 threads = 4 waves; each wave computes a 16x48 strip via WMMA f32 16x16x4.
// A tile (the 1 GiB stream) staged into LDS with async b128 copies (ASYNCcnt).
// ---------------------------------------------------------------------------
#define TM 64
#define TK 32

__global__ void __launch_bounds__(128) gemm_x_w1(
    const float* __restrict__ A, const float* __restrict__ B,
    float* __restrict__ Cout, int M, int K, int ncols)
{
    __shared__ __align__(16) float As[TM][TK + 4];   // stride 36 floats (144 B)
    __shared__ __align__(16) float Bs[TK][52];       // 48 cols zero-padded, stride 52

    const int tid  = threadIdx.x;
    const int wave = tid >> 5;
    const int lane = tid & 31;
    const int half = lane >> 4;        // 0: K-pair {0,1}; 1: K-pair {2,3}
    const int m16  = lane & 15;
    const long long rowBlock = (long long)blockIdx.x * TM;

    v8f acc0 = {}; v8f acc1 = {}; v8f acc2 = {};

    for (int k0 = 0; k0 < K; k0 += TK) {
        // --- A tile: 64x32 floats = 512 x b128, 4 async copies per thread ---
        #pragma unroll
        for (int i = 0; i < 4; ++i) {
            int idx = tid + i * 128;          // 0..511
            int r   = idx >> 3;               // 8 x b128 per row
            int c4  = (idx & 7) << 2;
            unsigned goff = (unsigned)(((rowBlock + r) * (long long)K + k0 + c4)
                                       * sizeof(float));
            // low 32 bits of a flat shared pointer == wave-relative LDS offset
            unsigned loff = (unsigned)(uintptr_t)&As[r][c4];
            async_copy_b128(loff, goff, A);
        }
        // --- B tile: 32x48, zero-pad cols >= ncols (tiny, L2-resident) ---
        #pragma unroll
        for (int i = 0; i < 12; ++i) {
            int idx = tid + i * 128;          // 0..1535
            int r = idx / 48;
            int c = idx - r * 48;
            Bs[r][c] = (c < ncols) ? B[(long long)(k0 + r) * ncols + c] : 0.0f;
        }
        if (k0 + TK < K) {  // warm L2 for the next A tile (global_prefetch_b8)
            __builtin_prefetch(A + (rowBlock + (tid >> 2)) * (long long)K + k0 + TK, 0, 1);
        }
        wait_asynccnt0();   // this wave's async LDS writes have landed
        __syncthreads();    // all waves' tile fills visible

        const int am = (wave << 4) + m16;
        #pragma unroll
        for (int kk = 0; kk < TK; kk += 4) {
            const int kr = kk + (half << 1);
            v2f a;  a.x  = As[am][kr];          a.y  = As[am][kr + 1];
            v2f b0; b0.x = Bs[kr][m16];         b0.y = Bs[kr + 1][m16];
            v2f b1; b1.x = Bs[kr][16 + m16];    b1.y = Bs[kr + 1][16 + m16];
            v2f b2; b2.x = Bs[kr][32 + m16];    b2.y = Bs[kr + 1][32 + m16];
            acc0 = wmma_f32_16x16x4(a, b0, acc0);
            acc1 = wmma_f32_16x16x4(a, b1, acc1);
            acc2 = wmma_f32_16x16x4(a, b2, acc2);
        }
        __syncthreads();
    }

    // C/D layout: VGPR v -> row v (lanes 0-15) / row v+8 (lanes 16-31)
    const long long rw = rowBlock + (wave << 4) + (half << 3);
    #pragma unroll
    for (int v = 0; v < 8; ++v) {
        const long long r = rw + v;
        Cout[r * ncols + m16] = acc0[v];                        // cols 0..15 always valid
        if (16 + m16 < ncols) Cout[r * ncols + 16 + m16] = acc1[v];
        if (32 + m16 < ncols) Cout[r * ncols + 32 + m16] = acc2[v];
    }
}

// ---------------------------------------------------------------------------
// Small-K GEMM (K multiple of 4, lda == K, ncols <= 48): layers 2 and 3.
// Each wave computes a 16-row strip; fragments loaded directly (tiny traffic).
// ---------------------------------------------------------------------------
__global__ void __launch_bounds__(128) gemm_small(
    const float* __restrict__ A, const float* __restrict__ B,
    float* __restrict__ Cout, int M, int K, int ncols)
{
    const int tid  = threadIdx.x;
    const int wave = tid >> 5;
    const int lane = tid & 31;
    const int half = lane >> 4;
    const int m16  = lane & 15;
    const long long row0 = (((long long)blockIdx.x << 2) + wave) << 4;
    if (row0 >= M) return;   // wave-uniform: EXEC stays all-ones for live waves

    v8f acc0 = {}; v8f acc1 = {}; v8f acc2 = {};
    const float* Arow = A + (row0 + m16) * (long long)K;
    const bool v1 = (16 + m16) < ncols;
    const bool v2 = (32 + m16) < ncols;

    for (int kk = 0; kk < K; kk += 4) {
        const int kr = kk + (half << 1);
        v2f a;  a.x = Arow[kr]; a.y = Arow[kr + 1];
        const float* B0 = B + (long long)kr * ncols;
        const float* B1 = B0 + ncols;
        v2f b0; b0.x = B0[m16];                   b0.y = B1[m16];
        v2f b1; b1.x = v1 ? B0[16 + m16] : 0.0f;  b1.y = v1 ? B1[16 + m16] : 0.0f;
        v2f b2; b2.x = v2 ? B0[32 + m16] : 0.0f;  b2.y = v2 ? B1[32 + m16] : 0.0f;
        acc0 = wmma_f32_16x16x4(a, b0, acc0);
        acc1 = wmma_f32_16x16x4(a, b1, acc1);
        acc2 = wmma_f32_16x16x4(a, b2, acc2);
    }

    const long long rw = row0 + (half << 3);
    #pragma unroll
    for (int v = 0; v < 8; ++v) {
        const long long r = rw + v;
        Cout[r * ncols + m16] = acc0[v];
        if (v1) Cout[r * ncols + 16 + m16] = acc1[v];
        if (v2) Cout[r * ncols + 32 + m16] = acc2[v];
    }
}

// ---------------------------------------------------------------------------
// Edge aggregation: agg[dst] += h[src] * dis[src]*dis[dst]   (fp32 HW atomics)
// ---------------------------------------------------------------------------
__global__ void aggregate_edges(
    const float* __restrict__ h, const int* __restrict__ src,
    const int* __restrict__ dst, const float* __restrict__ dis,
    float* __restrict__ agg, int E, int F)
{
    int e = blockIdx.x * blockDim.x + threadIdx.x;
    if (e >= E) return;
    const int s = src[e];
    const int d = dst[e];
    const float w = dis[s] * dis[d];
    const float* hs = h + (long long)s * F;
    float* ad = agg + (long long)d * F;
    for (int f = 0; f < F; ++f)
        unsafeAtomicAdd(&ad[f], hs[f] * w);
}

// out = act( agg + h * dis^2 (self-loop) + bias );  safe in-place (out == h)
__global__ void bias_act(
    const float* __restrict__ agg, const float* __restrict__ h,
    const float* __restrict__ dis, const float* __restrict__ bias,
    float* __restrict__ out, int N, int F, int do_relu)
{
    int i = blockIdx.x * blockDim.x + threadIdx.x;
    if (i >= N * F) return;
    const int n = i / F;
    const int f = i - n * F;
    const float di = dis[n];
    float v = agg[i] + h[i] * di * di + bias[f];
    out[i] = do_relu ? fmaxf(v, 0.0f) : v;
}

__global__ void softmax_rows(const float* __restrict__ in, float* __restrict__ out,
                             int N, int C)
{
    int n = blockIdx.x * blockDim.x + threadIdx.x;
    if (n >= N) return;
    const float* r = in + (long long)n * C;
    float* o = out + (long long)n * C;
    float mx = -3.402823466e38f;
    for (int c = 0; c < C; ++c) mx = fmaxf(mx, r[c]);
    float s = 0.0f;
    for (int c = 0; c < C; ++c) { float e = __expf(r[c] - mx); o[c] = e; s += e; }
    const float inv = 1.0f / s;
    for (int c = 0; c < C; ++c) o[c] *= inv;
}

// ---------------------------------------------------------------------------
// Host launcher
// ---------------------------------------------------------------------------
extern "C" void kernel_launch(void* const* d_in, const int* in_sizes, int n_in,
                              void* d_out, int out_size, void* d_ws, size_t ws_size,
                              hipStream_t stream)
{
    const float* x  = (const float*)d_in[0];
    const int*   ei = (const int*)d_in[1];
    const float* W1 = (const float*)d_in[2];
    const float* b1 = (const float*)d_in[3];
    const float* W2 = (const float*)d_in[4];
    const float* b2 = (const float*)d_in[5];
    const float* W3 = (const float*)d_in[6];
    const float* b3 = (const float*)d_in[7];
    float* out = (float*)d_out;

    const int H = in_sizes[3];           // 36
    const int C = in_sizes[7];           // 16
    const int E = in_sizes[1] / 2;       // 524288
    const int N = in_sizes[2] / H;       // 16384
    const int K = in_sizes[0] / N;       // 16384 (feature dim)

    const int* src = ei;        // edge_index[0]
    const int* dst = ei + E;    // edge_index[1]

    // Carve workspace (256B aligned): ~8.3 MB total
    uintptr_t p = ((uintptr_t)d_ws + 255) & ~(uintptr_t)255;
    auto carve = [&](size_t nElem) {
        float* q = (float*)p;
        p += ((nElem * sizeof(float) + 255) / 256) * 256;
        return q;
    };
    float* deg = carve((size_t)N);
    float* dis = carve((size_t)N);
    float* h1  = carve((size_t)N * H);
    float* h2  = carve((size_t)N * H);
    float* h3  = carve((size_t)N * C);
    float* agg = carve((size_t)N * H);
    (void)ws_size; (void)n_in; (void)out_size;

    const int T = 256;
    const long long NH = (long long)N * H;
    const long long NC = (long long)N * C;

    // --- normalization ---
    zero_f32   <<<(N + T - 1) / T, T, 0, stream>>>(deg, N);
    count_deg  <<<(E + T - 1) / T, T, 0, stream>>>(dst, deg, E);
    compute_dis<<<(N + T - 1) / T, T, 0, stream>>>(deg, dis, N);

    // --- layer 1: h1 = x @ W1 ; aggregate ; +b1 ; relu ---
    gemm_x_w1<<<N / TM, 128, 0, stream>>>(x, W1, h1, N, K, H);
    zero_f32<<<(int)((NH + T - 1) / T), T, 0, stream>>>(agg, NH);
    aggregate_edges<<<(E + T - 1) / T, T, 0, stream>>>(h1, src, dst, dis, agg, E, H);
    bias_act<<<(int)((NH + T - 1) / T), T, 0, stream>>>(agg, h1, dis, b1, h1, N, H, 1);

    // --- layer 2 ---
    gemm_small<<<(N / 16 + 3) / 4, 128, 0, stream>>>(h1, W2, h2, N, H, H);
    zero_f32<<<(int)((NH + T - 1) / T), T, 0, stream>>>(agg, NH);
    aggregate_edges<<<(E + T - 1) / T, T, 0, stream>>>(h2, src, dst, dis, agg, E, H);
    bias_act<<<(int)((NH + T - 1) / T), T, 0, stream>>>(agg, h2, dis, b2, h2, N, H, 1);

    // --- layer 3 (no relu) + softmax ---
    gemm_small<<<(N / 16 + 3) / 4, 128, 0, stream>>>(h2, W3, h3, N, H, C);
    zero_f32<<<(int)((NC + T - 1) / T), T, 0, stream>>>(agg, NC);
    aggregate_edges<<<(E + T - 1) / T, T, 0, stream>>>(h3, src, dst, dis, agg, E, C);
    bias_act<<<(int)((NC + T - 1) / T), T, 0, stream>>>(agg, h3, dis, b3, h3, N, C, 0);

    softmax_rows<<<(N + T - 1) / T, T, 0, stream>>>(h3, out, N, C);
}